// FourierBlock_45423574122667
// MI455X (gfx1250) — compile-verified
//
#include <hip/hip_runtime.h>

// ---------------------------------------------------------------------------
// FourierBlock for MI455X (gfx1250): truncated-DFT formulation.
//   fwd:  X[bh][e][0:64]=sum_t x*cos, [64:128]=sum_t x*sin   (x_ft = Xc - i*Xs)
//   mix:  O = (Xc - i Xs)(w1 + i w2); C[e][m]=a_m*Or, C[e][64+m]=-a_m*Oi,
//         a_m = (m==0?1:2)/L   (irfft zero-padded to L/2+1)
//   inv:  out[e][t] = sum_k C[e][k] * Ft[k][t]   (Ft = [cos;sin] transposed)
// Heavy GEMMs use V_WMMA_F32_16X16X4_F32 (fp32 in/accum for spectral accuracy).
// ---------------------------------------------------------------------------

typedef float v2f __attribute__((ext_vector_type(2)));
typedef float v8f __attribute__((ext_vector_type(8)));

#define B_    8
#define L_    4096
#define H_    8
#define E_    64
#define M_    64
#define HE    (H_*E_)     // 512
#define NCOL  128         // cos||sin columns

// workspace layout (floats)
#define WS_F    0          // F  [4096][128]
#define WS_FT   524288     // Ft [128][4096]
#define WS_X    1048576    // X  [B*H][64][128]
#define WS_C    1572864    // C  [B*H][64][128]

// ---------------------------------------------------------------- twiddles --
__global__ __launch_bounds__(256) void k_twiddle(float* __restrict__ F,
                                                 float* __restrict__ Ft) {
  int idx = blockIdx.x * 256 + threadIdx.x;    // 262144 = 4096*64
  int t = idx >> 6;
  int m = idx & 63;
  int ph = (t * m) & (L_ - 1);                 // periodic -> keep arg small
  float ang = (float)ph * (6.28318530717958647692f / (float)L_);
  float s, c;
  sincosf(ang, &s, &c);
  F[t * NCOL + m]      = c;
  F[t * NCOL + 64 + m] = s;
  Ft[m * L_ + t]        = c;
  Ft[(64 + m) * L_ + t] = s;
}

// ------------------------------------------------------------- forward DFT --
// grid: 1024 blocks = 64 bh * 4 etile * 4 npair ; 256 thr = 8 waves.
// wave w: ntl = w>>2 (two 16-col tiles), kq = w&3 (K quarter, 1024 t each).
// Deterministic cross-wave K reduction through LDS (no float atomics).
__global__ __launch_bounds__(256) void k_dft_fwd(const float* __restrict__ q,
                                                 const float* __restrict__ F,
                                                 float* __restrict__ X) {
  __shared__ float red[8][256];
  int wave = threadIdx.x >> 5;
  int lane = threadIdx.x & 31;
  int bh    = blockIdx.x >> 4;          // 0..63
  int etile = (blockIdx.x >> 2) & 3;    // 0..3
  int npair = blockIdx.x & 3;           // 0..3
  int ntl = wave >> 2;                  // 0..1
  int kq  = wave & 3;                   // 0..3
  int ntile = npair * 2 + ntl;          // 0..7
  int row = lane & 15;
  int hi  = lane >> 4;
  int b = bh >> 3, h = bh & 7;

  // A[e][t] = q[b][t][h][e] ; lane owns e = etile*16+row, K pair = 2*hi
  const float* qa = q + (size_t)b * L_ * HE + (size_t)h * E_ + etile * 16 + row;
  // B[t][n] = F[t][ntile*16+row]
  const float* fb = F + ntile * 16 + row;

  v8f acc = {};
  int t_end = kq * 1024 + 1024;
#pragma unroll 4
  for (int t0 = kq * 1024; t0 < t_end; t0 += 4) {
    int ta = t0 + 2 * hi;
    v2f a, bb;
    a.x  = qa[(size_t)ta * HE];
    a.y  = qa[(size_t)(ta + 1) * HE];
    bb.x = fb[(size_t)ta * NCOL];
    bb.y = fb[(size_t)(ta + 1) * NCOL];
    acc = __builtin_amdgcn_wmma_f32_16x16x4_f32(false, a, false, bb,
                                                (short)0, acc, false, false);
  }
  // dump partial tile: pos = m_local*16 + n
#pragma unroll
  for (int j = 0; j < 8; ++j) {
    int ml = j + 8 * hi;
    red[wave][ml * 16 + row] = acc[j];
  }
  __syncthreads();
  // reduce 4 K-quarters per ntl, fixed order -> deterministic
#pragma unroll
  for (int r = 0; r < 2; ++r) {
    int idx = threadIdx.x + r * 256;        // 0..511
    int tl  = idx >> 8;                     // which ntl
    int pos = idx & 255;
    float v = red[tl * 4 + 0][pos] + red[tl * 4 + 1][pos] +
              red[tl * 4 + 2][pos] + red[tl * 4 + 3][pos];
    int ml = pos >> 4, n = pos & 15;
    int er  = etile * 16 + ml;
    int col = (npair * 2 + tl) * 16 + n;
    X[((size_t)bh * E_ + er) * NCOL + col] = v;
  }
}

// ---------------------------------------------------------------- mode mix --
// thread per (b,h,o,m): 262144 threads, 64 complex MACs each (134 MFLOP).
__global__ __launch_bounds__(256) void k_mix(const float* __restrict__ X,
                                             const float* __restrict__ w1,
                                             const float* __restrict__ w2,
                                             float* __restrict__ C) {
  int tid = blockIdx.x * 256 + threadIdx.x;
  int m  = tid & 63;
  int o  = (tid >> 6) & 63;
  int hh = (tid >> 12) & 7;
  int b  = tid >> 15;
  const float* xp = X + ((size_t)(b * H_ + hh) * E_) * NCOL + m;      // + i*NCOL
  const float* wr = w1 + ((size_t)hh * E_ * E_ + o) * M_ + m;         // + i*E_*M_
  const float* wi = w2 + ((size_t)hh * E_ * E_ + o) * M_ + m;
  float Or = 0.f, Oi = 0.f;
#pragma unroll 4
  for (int i = 0; i < E_; ++i) {
    float xc = xp[(size_t)i * NCOL];
    float xs = xp[(size_t)i * NCOL + 64];
    float wrr = wr[(size_t)i * E_ * M_];
    float wii = wi[(size_t)i * E_ * M_];
    // x_ft = xc - i*xs  ->  (xc*wr + xs*wi) + i*(xc*wi - xs*wr)
    Or += xc * wrr + xs * wii;
    Oi += xc * wii - xs * wrr;
  }
  float am = (m == 0 ? 1.0f : 2.0f) * (1.0f / (float)L_);
  float* cf = C + ((size_t)(b * H_ + hh) * E_ + o) * NCOL;
  cf[m]      =  am * Or;   // multiplies cos rows of Ft
  cf[64 + m] = -am * Oi;   // multiplies sin rows of Ft
}

// ------------------------------------------------------------- inverse DFT --
// grid: 8192 blocks = 64 bh * 128 t-slabs(32t) ; 8 waves = 4 etile x 2 ntloc.
// WMMA K=128, then LDS transpose so stores are contiguous float4 in e.
__global__ __launch_bounds__(256) void k_dft_inv(const float* __restrict__ C,
                                                 const float* __restrict__ Ft,
                                                 float* __restrict__ out) {
  __shared__ float lds[32 * 68];          // [t][e], pitch 68 (pad: banks+f4 align)
  int wave = threadIdx.x >> 5;
  int lane = threadIdx.x & 31;
  int bh = blockIdx.x >> 7;
  int nb = blockIdx.x & 127;
  int tbase = nb * 32;
  int etile = wave >> 1;
  int ntloc = wave & 1;
  int row = lane & 15;
  int hi  = lane >> 4;
  int e = etile * 16 + row;
  int t = tbase + ntloc * 16 + row;

  const v2f* ca  = (const v2f*)(C + ((size_t)bh * E_ + e) * NCOL);  // pair idx
  const float* fb = Ft + t;

  v8f acc = {};
#pragma unroll
  for (int k0 = 0; k0 < NCOL; k0 += 4) {
    v2f a = ca[(k0 >> 1) + hi];           // C[e][k0+2hi], C[e][k0+2hi+1]
    v2f bb;
    bb.x = fb[(size_t)(k0 + 2 * hi) * L_];
    bb.y = fb[(size_t)(k0 + 2 * hi + 1) * L_];
    acc = __builtin_amdgcn_wmma_f32_16x16x4_f32(false, a, false, bb,
                                                (short)0, acc, false, false);
  }
  int tl = ntloc * 16 + row;
#pragma unroll
  for (int j = 0; j < 8; ++j) {
    int er = etile * 16 + j + 8 * hi;
    lds[tl * 68 + er] = acc[j];
  }
  __syncthreads();

  int b = bh >> 3, h = bh & 7;
  float* ob = out + ((size_t)b * L_ + tbase) * HE + (size_t)h * E_;
#pragma unroll
  for (int r = 0; r < 2; ++r) {
    int idx = threadIdx.x + r * 256;      // 512 float4s = 32t * 64e
    int tt = idx >> 4;
    int e0 = (idx & 15) << 2;
    float4 v;
    v.x = lds[tt * 68 + e0 + 0];
    v.y = lds[tt * 68 + e0 + 1];
    v.z = lds[tt * 68 + e0 + 2];
    v.w = lds[tt * 68 + e0 + 3];
    *(float4*)(ob + (size_t)tt * HE + e0) = v;
  }
}

// ---------------------------------------------------------------------------
extern "C" void kernel_launch(void* const* d_in, const int* in_sizes, int n_in,
                              void* d_out, int out_size, void* d_ws, size_t ws_size,
                              hipStream_t stream) {
  const float* q  = (const float*)d_in[0];
  const float* w1 = (const float*)d_in[3];
  const float* w2 = (const float*)d_in[4];
  float* out = (float*)d_out;

  float* wsf = (float*)d_ws;              // needs 8 MB
  float* F  = wsf + WS_F;
  float* Ft = wsf + WS_FT;
  float* X  = wsf + WS_X;
  float* C  = wsf + WS_C;

  k_twiddle<<<1024, 256, 0, stream>>>(F, Ft);
  k_dft_fwd<<<1024, 256, 0, stream>>>(q, F, X);
  k_mix    <<<1024, 256, 0, stream>>>(X, w1, w2, C);
  k_dft_inv<<<8192, 256, 0, stream>>>(C, Ft, out);
}